// VectorQuantizer_18159121728134
// MI455X (gfx1250) — compile-verified
//
#include <hip/hip_runtime.h>

typedef float v2f __attribute__((ext_vector_type(2)));
typedef float v4f __attribute__((ext_vector_type(4)));
typedef float v8f __attribute__((ext_vector_type(8)));

#define D_DIM 64
#define KCODES 1024
#define CHUNK 32               // codebook rows staged in LDS per iteration
#define NCHUNK (KCODES / CHUNK)
#define LDS_STRIDE 68          // padded row stride (floats): 16B-aligned rows, bank-spread
#define BUF_FLOATS (CHUNK * LDS_STRIDE)
#define ROWS_PER_WAVE 16
#define WAVES_PER_BLOCK 8
#define ROWS_PER_BLOCK (ROWS_PER_WAVE * WAVES_PER_BLOCK)

__global__ void vq_init_kernel(float* lossSlot) {
    if (threadIdx.x == 0) *lossSlot = 0.0f;
}

// Issue this thread's share (2 x b128) of one codebook chunk as async
// global->LDS copies (GVS mode: lds-addr VGPR, 32-bit byte-offset VGPR,
// 64-bit SGPR base). Tracked by ASYNCcnt.
__device__ __forceinline__ void async_stage_chunk(const float* __restrict__ E,
                                                  unsigned sEbyteBase,
                                                  int buf, int ch, int tid) {
    #pragma unroll
    for (int k = 0; k < 2; ++k) {
        const int i    = tid + k * 256;   // 0..511 : 512 b128 transfers per chunk
        const int row  = i >> 4;          // 16 transfers (64 floats) per row
        const int part = i & 15;
        const unsigned goff = (unsigned)(((ch * CHUNK + row) * D_DIM + part * 4) * 4);
        const unsigned loff = sEbyteBase +
            (unsigned)((buf * BUF_FLOATS + row * LDS_STRIDE + part * 4) * 4);
        asm volatile("global_load_async_to_lds_b128 %0, %1, %2"
                     :: "v"(loff), "v"(goff), "s"(E)
                     : "memory");
    }
}

__global__ __launch_bounds__(256) void vq_kernel(
    const float* __restrict__ X,      // [N, 64]
    const float* __restrict__ E,      // [1024, 64]
    float* __restrict__ outQ,         // [N, 64]
    float* __restrict__ outIdx,       // [N] (index values, stored as f32)
    float* __restrict__ outLoss,      // [1]
    float lossScale)                  // 1.25 / (N*64)
{
    __shared__ float sE[2 * BUF_FLOATS];   // double-buffered codebook chunks
    __shared__ float sN[KCODES];           // codebook squared norms

    const int tid  = threadIdx.x;
    const int lane = tid & 31;
    const int wave = tid >> 5;
    const int half = lane >> 4;            // which 16-lane half of the wave
    const int l16  = lane & 15;

    const unsigned sEbase = (unsigned)(size_t)(void*)&sE[0];

    // kick off chunk 0 before doing anything else
    async_stage_chunk(E, sEbase, 0, 0, tid);

    // ---- precompute codebook squared norms into LDS (overlaps async copy) ----
    for (int c = tid; c < KCODES; c += 256) {
        const v4f* row = (const v4f*)(E + (size_t)c * D_DIM);
        float s = 0.0f;
        #pragma unroll
        for (int i = 0; i < D_DIM / 4; ++i) {
            v4f v = row[i];
            s += v.x * v.x + v.y * v.y + v.z * v.z + v.w * v.w;
        }
        sN[c] = s;
    }

    const int m0 = blockIdx.x * ROWS_PER_BLOCK + wave * ROWS_PER_WAVE;

    // ---- preload this wave's 16 X rows in WMMA f32 16x16x4 A-layout ----
    // step s covers d = 4s..4s+3: lanes 0-15 hold (K=4s, 4s+1), lanes 16-31 (4s+2, 4s+3)
    v2f a[16];
    {
        const float* xrow = X + (size_t)(m0 + l16) * D_DIM;
        #pragma unroll
        for (int s = 0; s < 16; ++s)
            a[s] = *(const v2f*)(xrow + 4 * s + 2 * half);
    }

    float minv[8];
    int   mini[8];
    #pragma unroll
    for (int j = 0; j < 8; ++j) { minv[j] = 3.4e38f; mini[j] = 0; }

    for (int ch = 0; ch < NCHUNK; ++ch) {
        const int buf = ch & 1;

        // stage next chunk into the other buffer, then wait for *this* chunk:
        // async completions are in-order, so asynccnt<=2 (the 2 just issued)
        // implies chunk `ch` has fully landed in LDS.
        if (ch + 1 < NCHUNK) {
            async_stage_chunk(E, sEbase, buf ^ 1, ch + 1, tid);
            asm volatile("s_wait_asynccnt 0x2" ::: "memory");
        } else {
            asm volatile("s_wait_asynccnt 0x0" ::: "memory");
        }
        __syncthreads();

        // ---- two independent 16-code tiles -> two WMMA accumulator chains ----
        const float* bbase = sE + buf * BUF_FLOATS;
        const int   n0  = ch * CHUNK + l16;       // tile-0 code id for this lane
        const float en0 = sN[n0];
        const float en1 = sN[n0 + 16];
        const float* b0 = bbase + l16 * LDS_STRIDE + 2 * half;
        const float* b1 = b0 + 16 * LDS_STRIDE;

        v8f acc0 = {};
        v8f acc1 = {};
        #pragma unroll
        for (int s = 0; s < 16; ++s) {
            v2f bb0 = *(const v2f*)(b0 + 4 * s);
            v2f bb1 = *(const v2f*)(b1 + 4 * s);
            acc0 = __builtin_amdgcn_wmma_f32_16x16x4_f32(
                false, a[s], false, bb0, (short)0, acc0, false, false);
            acc1 = __builtin_amdgcn_wmma_f32_16x16x4_f32(
                false, a[s], false, bb1, (short)0, acc1, false, false);
        }

        // distance (minus the row-constant ||x||^2): ||e||^2 - 2 x.e
        // tile 0 tested before tile 1 with strict < => lowest index wins ties.
        #pragma unroll
        for (int j = 0; j < 8; ++j) {
            const float v0 = en0 - 2.0f * acc0[j];
            if (v0 < minv[j]) { minv[j] = v0; mini[j] = n0; }
            const float v1 = en1 - 2.0f * acc1[j];
            if (v1 < minv[j]) { minv[j] = v1; mini[j] = n0 + 16; }
        }
        __syncthreads();   // all waves done reading `buf` before it is refilled
    }

    // ---- argmin across the 16 lanes of each half (C-layout: VGPR j = rows j, j+8) ----
    #pragma unroll
    for (int j = 0; j < 8; ++j) {
        #pragma unroll
        for (int off = 8; off >= 1; off >>= 1) {
            float ov = __shfl_xor(minv[j], off, 32);
            int   oi = __shfl_xor(mini[j], off, 32);
            if (ov < minv[j] || (ov == minv[j] && oi < mini[j])) {
                minv[j] = ov; mini[j] = oi;
            }
        }
    }

    // ---- gather code rows, write quantized + indices, accumulate loss ----
    float lacc = 0.0f;
    #pragma unroll
    for (int r = 0; r < 16; ++r) {
        const int idx = __shfl(mini[r & 7], (r >> 3) << 4, 32);
        const int m   = m0 + r;
        v2f e2 = *(const v2f*)(E + (size_t)idx * D_DIM + 2 * lane);
        v2f x2 = *(const v2f*)(X + (size_t)m   * D_DIM + 2 * lane);
        const float d0 = e2.x - x2.x;
        const float d1 = e2.y - x2.y;
        lacc += d0 * d0 + d1 * d1;
        *(v2f*)(outQ + (size_t)m * D_DIM + 2 * lane) = e2;   // straight-through == quantized
        if (lane == 0) outIdx[m] = (float)idx;
    }

    #pragma unroll
    for (int off = 16; off >= 1; off >>= 1)
        lacc += __shfl_xor(lacc, off, 32);
    if (lane == 0)
        atomicAdd(outLoss, lacc * lossScale);
}

extern "C" void kernel_launch(void* const* d_in, const int* in_sizes, int n_in,
                              void* d_out, int out_size, void* d_ws, size_t ws_size,
                              hipStream_t stream) {
    const float* X = (const float*)d_in[0];   // inputs  [64,32,32,64] f32
    const float* E = (const float*)d_in[1];   // codebook [1024,64]   f32

    const int nElem = in_sizes[0];            // 4194304
    const int nRows = nElem / D_DIM;          // 65536

    float* outQ    = (float*)d_out;
    float* outIdx  = outQ + nElem;
    float* outLoss = outIdx + nRows;

    vq_init_kernel<<<1, 32, 0, stream>>>(outLoss);

    const int blocks = nRows / ROWS_PER_BLOCK; // 512
    const float lossScale = 1.25f / (float)nElem;
    vq_kernel<<<blocks, 256, 0, stream>>>(X, E, outQ, outIdx, outLoss, lossScale);
}